// PointNetNetwork_34849364640187
// MI455X (gfx1250) — compile-verified
//
#include <hip/hip_runtime.h>

// PointNet++ (SA/SA/summit/FP/FP/cls) forward for MI455X (gfx1250, wave32).
// GEMMs run on V_WMMA_F32_16X16X4_F32 (fp32 matrix pipe, matches reference
// precision). Weights are pre-transposed to [N,Kd] so both A and B fragments
// load contiguously along K (b64 loads, no strided access), and the K main
// loop is unguarded (no exec-mask toggling around loads). FPS is a single
// 32-wave block with LDS-resident min-distance array and wave32 shuffle
// argmax. Radius-graph top-32 uses per-thread insertion lists interleaved in
// LDS. No atomics -> bit-deterministic.
//
// Input flattening assumption (cannot be validated in compile-only loop):
//   d_in[0]=x[8192,4] d_in[1]=pos[8192,3] d_in[2]=batch (unused), then params
//   as JAX pytree leaves: cls(W,b | W,b), down0(W,b,be,g | W,b,be,g),
//   down1(...), summit(W,b), up0(W,b,be,g), up1(W,b,be,g).
// Workspace requirement: ~248 MB.

#define KNN 32

typedef float v2f __attribute__((ext_vector_type(2)));
typedef float v8f __attribute__((ext_vector_type(8)));

// ---------------- radius graph: 32 nearest neighbors + radius mask ----------
__global__ void k_radius_topk(const float* __restrict__ pos, int Np, float r2,
                              int* __restrict__ nbr, int* __restrict__ msk)
{
  extern __shared__ float sm[];
  const int tid = threadIdx.x, bs = blockDim.x;
  float* sd = sm;                                  // [KNN][bs] interleaved
  int*   si = (int*)(sm + (size_t)KNN * bs);
  const int i = blockIdx.x * bs + tid;
  if (i >= Np) return;
  const float px = pos[3*i], py = pos[3*i+1], pz = pos[3*i+2];
#define BD(k) sd[(k)*bs + tid]
#define BI(k) si[(k)*bs + tid]
  for (int k = 0; k < KNN; ++k) { BD(k) = __builtin_inff(); BI(k) = 0; }
  float worst = __builtin_inff();
  for (int j = 0; j < Np; ++j) {
    if (j == i) continue;
    float dx = pos[3*j] - px, dy = pos[3*j+1] - py, dz = pos[3*j+2] - pz;
    float d = dx*dx + dy*dy + dz*dz;
    if (d < worst) {                               // stable insert (ties -> lower idx)
      int q = KNN - 1;
      while (q > 0 && BD(q-1) > d) { BD(q) = BD(q-1); BI(q) = BI(q-1); --q; }
      BD(q) = d; BI(q) = j;
      worst = BD(KNN-1);
    }
  }
  for (int k = 0; k < KNN; ++k) {
    nbr[(size_t)i*KNN + k] = BI(k);
    msk[(size_t)i*KNN + k] = (BD(k) < r2) ? 1 : 0;
  }
#undef BD
#undef BI
}

// ---------------- farthest point sampling (serial scan, one workgroup) ------
__global__ void k_fps(const float* __restrict__ pos, int Np, int n,
                      int* __restrict__ out)
{
  extern __shared__ float mind[];                  // [Np]
  __shared__ float rv[32]; __shared__ int ri[32]; __shared__ int s_last;
  const int tid = threadIdx.x, bs = blockDim.x;
  const int lane = tid & 31, wid = tid >> 5, nw = bs >> 5;
  for (int j = tid; j < Np; j += bs) mind[j] = __builtin_inff();
  __syncthreads();
  int last = 0;
  for (int it = 0; it < n; ++it) {
    if (tid == 0) out[it] = last;
    const float lx = pos[3*last], ly = pos[3*last+1], lz = pos[3*last+2];
    float bv = -__builtin_inff(); int bi_ = 0x7fffffff;
    for (int j = tid; j < Np; j += bs) {
      float dx = pos[3*j]-lx, dy = pos[3*j+1]-ly, dz = pos[3*j+2]-lz;
      float m = fminf(mind[j], dx*dx + dy*dy + dz*dz);
      mind[j] = m;
      if (m > bv || (m == bv && j < bi_)) { bv = m; bi_ = j; }
    }
    for (int o = 16; o; o >>= 1) {                 // wave32 argmax
      float ov = __shfl_down(bv, o, 32);
      int   oi = __shfl_down(bi_, o, 32);
      if (ov > bv || (ov == bv && oi < bi_)) { bv = ov; bi_ = oi; }
    }
    if (lane == 0) { rv[wid] = bv; ri[wid] = bi_; }
    __syncthreads();
    if (wid == 0) {
      bv  = (lane < nw) ? rv[lane] : -__builtin_inff();
      bi_ = (lane < nw) ? ri[lane] : 0x7fffffff;
      for (int o = 16; o; o >>= 1) {
        float ov = __shfl_down(bv, o, 32);
        int   oi = __shfl_down(bi_, o, 32);
        if (ov > bv || (ov == bv && oi < bi_)) { bv = ov; bi_ = oi; }
      }
      if (lane == 0) s_last = bi_;
    }
    __syncthreads();
    last = s_last;
  }
}

// ---------------- gather neighbor features [x[j], pos[j]-pos[i]] ------------
__global__ void k_build_feat(const float* __restrict__ x, int Cx,
                             const float* __restrict__ pos,
                             const int* __restrict__ nbr, int Np,
                             float* __restrict__ feat)
{
  long t = (long)blockIdx.x * blockDim.x + threadIdx.x;
  if (t >= (long)Np * KNN) return;
  int i = (int)(t / KNN);
  int j = nbr[t];
  float* f = feat + t * (Cx + 3);
  for (int c = 0; c < Cx; ++c) f[c] = x[(long)j*Cx + c];
  f[Cx+0] = pos[3*j+0] - pos[3*i+0];
  f[Cx+1] = pos[3*j+1] - pos[3*i+1];
  f[Cx+2] = pos[3*j+2] - pos[3*i+2];
}

// ---------------- small weight transpose: W[Kd,N] -> Wt[N,Kd] ---------------
__global__ void k_transpose(const float* __restrict__ W, int Kd, int N,
                            float* __restrict__ Wt)
{
  long t = (long)blockIdx.x * blockDim.x + threadIdx.x;
  if (t >= (long)Kd * N) return;
  int k = (int)(t / N), n2 = (int)(t % N);
  Wt[(long)n2*Kd + k] = W[t];
}

// ---------------- fp32 WMMA GEMM: C[M,N] = A[M,Kd] @ W[Kd,N] + bias ---------
// Wt is the transposed weight [N,Kd]. One wave per 16x16 output tile; K
// stepped by 4 via V_WMMA_F32_16X16X4_F32. Main loop (K multiple of 16) is
// fully unguarded: contiguous A/B fragment loads, 4 WMMAs, 2 accumulators to
// break the D->C dependency chain. Guarded tail only for Kd=7/131.
// Requires M%16==0 and N%16==0 (true for every layer here).
__global__ void k_gemm_wmma(const float* __restrict__ A, const float* __restrict__ Wt,
                            const float* __restrict__ bias, float* __restrict__ C,
                            int M, int Kd, int N, int relu)
{
  const int lane = threadIdx.x & 31;
  const int wave = threadIdx.x >> 5;
  const long tilesN = N >> 4;
  const long tiles  = ((long)M >> 4) * tilesN;
  const long tile   = (long)blockIdx.x * (blockDim.x >> 5) + wave;
  if (tile >= tiles) return;                       // wave-uniform: EXEC stays full
  const int row0 = (int)((tile / tilesN) << 4);
  const int col0 = (int)((tile % tilesN) << 4);
  const int half = lane >> 4;                      // 0: K,K+1  1: K+2,K+3
  const int l16  = lane & 15;
  const int coln = col0 + l16;                     // B/C column for this lane
  const float* ap = A  + (long)(row0 + l16) * Kd + 2*half;  // A row, this lane
  const float* bp = Wt + (long)coln * Kd + 2*half;          // W column, this lane
  v8f acc0 = {}, acc1 = {};
  const int kmain = Kd & ~15;
  int k0 = 0;
  for (; k0 < kmain; k0 += 16) {                   // unguarded hot loop
    v2f a0, b0, a1, b1, a2, b2, a3, b3;
    a0.x = ap[k0+0];  a0.y = ap[k0+1];   b0.x = bp[k0+0];  b0.y = bp[k0+1];
    a1.x = ap[k0+4];  a1.y = ap[k0+5];   b1.x = bp[k0+4];  b1.y = bp[k0+5];
    a2.x = ap[k0+8];  a2.y = ap[k0+9];   b2.x = bp[k0+8];  b2.y = bp[k0+9];
    a3.x = ap[k0+12]; a3.y = ap[k0+13];  b3.x = bp[k0+12]; b3.y = bp[k0+13];
    acc0 = __builtin_amdgcn_wmma_f32_16x16x4_f32(false, a0, false, b0, (short)0, acc0, false, false);
    acc1 = __builtin_amdgcn_wmma_f32_16x16x4_f32(false, a1, false, b1, (short)0, acc1, false, false);
    acc0 = __builtin_amdgcn_wmma_f32_16x16x4_f32(false, a2, false, b2, (short)0, acc0, false, false);
    acc1 = __builtin_amdgcn_wmma_f32_16x16x4_f32(false, a3, false, b3, (short)0, acc1, false, false);
  }
  for (; k0 < Kd; k0 += 4) {                       // guarded tail (Kd=7,131 only)
    const int ka = k0 + 2*half;
    v2f a, b;
    a.x = (ka     < Kd) ? ap[k0]   : 0.f;
    a.y = (ka + 1 < Kd) ? ap[k0+1] : 0.f;
    b.x = (ka     < Kd) ? bp[k0]   : 0.f;
    b.y = (ka + 1 < Kd) ? bp[k0+1] : 0.f;
    acc0 = __builtin_amdgcn_wmma_f32_16x16x4_f32(false, a, false, b, (short)0, acc0, false, false);
  }
  const v8f acc = acc0 + acc1;
  const float bv = bias ? bias[coln] : 0.f;
#pragma unroll
  for (int r = 0; r < 8; ++r) {                    // D: VGPR r -> M = r + 8*half
    float v = acc[r] + bv;
    if (relu) v = fmaxf(v, 0.f);
    C[(long)(row0 + r + 8*half) * N + coln] = v;
  }
}

// ---------------- BatchNorm (train mode) stats: deterministic tree reduce ---
__global__ void k_bn_stats(const float* __restrict__ h, const int* __restrict__ mask,
                           long Mr, int C, float* __restrict__ stats)
{
  const int c = blockIdx.x, tid = threadIdx.x;
  float s = 0.f, q = 0.f, cn = 0.f;
  for (long r = tid; r < Mr; r += blockDim.x) {
    float w = mask ? (mask[r] ? 1.f : 0.f) : 1.f;
    float v = h[r*C + c];
    s += w*v; q += w*v*v; cn += w;
  }
  __shared__ float rs[256], rq[256], rc[256];
  rs[tid] = s; rq[tid] = q; rc[tid] = cn;
  __syncthreads();
  for (int o = 128; o; o >>= 1) {
    if (tid < o) { rs[tid]+=rs[tid+o]; rq[tid]+=rq[tid+o]; rc[tid]+=rc[tid+o]; }
    __syncthreads();
  }
  if (tid == 0) {
    stats[c] = rs[0]; stats[C + c] = rq[0];
    if (c == 0) stats[2*C] = rc[0];
  }
}

__global__ void k_bn_apply_relu(float* __restrict__ h, long Mr, int C,
                                const float* __restrict__ stats,
                                const float* __restrict__ g,
                                const float* __restrict__ be)
{
  long t = (long)blockIdx.x * blockDim.x + threadIdx.x;
  if (t >= Mr * C) return;
  int c = (int)(t % C);
  float cnt = fmaxf(stats[2*C], 1.f);
  float m = stats[c] / cnt;
  float v = fmaxf(stats[C + c] / cnt - m*m, 0.f);  // E[x^2]-m^2 over masked set
  float y = (h[t] - m) * rsqrtf(v + 1e-5f) * g[c] + be[c];
  h[t] = fmaxf(y, 0.f);
}

// ---------------- masked max aggregation over K neighbors -------------------
__global__ void k_max_agg(const float* __restrict__ h, const int* __restrict__ mask,
                          int Np, int C, float* __restrict__ out)
{
  long t = (long)blockIdx.x * blockDim.x + threadIdx.x;
  if (t >= (long)Np * C) return;
  int i = (int)(t / C), c = (int)(t % C);
  float best = -__builtin_inff(); int any = 0;
  for (int k = 0; k < KNN; ++k)
    if (mask[(long)i*KNN + k]) { any = 1; best = fmaxf(best, h[((long)i*KNN + k)*C + c]); }
  out[t] = any ? best : 0.f;
}

// ---------------- row gather / 2-way concat ---------------------------------
__global__ void k_gather_rows(const float* __restrict__ src, int C,
                              const int* __restrict__ idx, int n,
                              float* __restrict__ dst)
{
  long t = (long)blockIdx.x * blockDim.x + threadIdx.x;
  if (t >= (long)n * C) return;
  int r = (int)(t / C), c = (int)(t % C);
  dst[t] = src[(long)idx[r]*C + c];
}

__global__ void k_concat2(const float* __restrict__ a, int Ca,
                          const float* __restrict__ b, int Cb,
                          int n, float* __restrict__ out)
{
  int Ct = Ca + Cb;
  long t = (long)blockIdx.x * blockDim.x + threadIdx.x;
  if (t >= (long)n * Ct) return;
  int r = (int)(t / Ct), c = (int)(t % Ct);
  out[t] = (c < Ca) ? a[(long)r*Ca + c] : b[(long)r*Cb + (c - Ca)];
}

// ---------------- inverse-d2-weighted 3-NN interpolation --------------------
__global__ void k_knn3_interp(const float* __restrict__ pd, int Nd,
                              const float* __restrict__ ps, int Ns,
                              const float* __restrict__ xs, int C,
                              float* __restrict__ out)
{
  int i = blockIdx.x * blockDim.x + threadIdx.x;
  if (i >= Nd) return;
  float px = pd[3*i], py = pd[3*i+1], pz = pd[3*i+2];
  float d0 = __builtin_inff(), d1 = d0, d2 = d0; int i0 = 0, i1 = 0, i2 = 0;
  for (int j = 0; j < Ns; ++j) {
    float dx = ps[3*j]-px, dy = ps[3*j+1]-py, dz = ps[3*j+2]-pz;
    float d = dx*dx + dy*dy + dz*dz;
    if (d < d0)      { d2=d1;i2=i1; d1=d0;i1=i0; d0=d;i0=j; }
    else if (d < d1) { d2=d1;i2=i1; d1=d;i1=j; }
    else if (d < d2) { d2=d;i2=j; }
  }
  float w0 = 1.f/fmaxf(d0,1e-16f), w1 = 1.f/fmaxf(d1,1e-16f), w2 = 1.f/fmaxf(d2,1e-16f);
  float inv = 1.f/(w0+w1+w2);
  for (int c = 0; c < C; ++c)
    out[(long)i*C + c] =
        (w0*xs[(long)i0*C+c] + w1*xs[(long)i1*C+c] + w2*xs[(long)i2*C+c]) * inv;
}

// ============================================================================
static inline int cdivl(long a, int b) { return (int)((a + b - 1) / b); }

extern "C" void kernel_launch(void* const* d_in, const int* in_sizes, int n_in,
                              void* d_out, int out_size, void* d_ws, size_t ws_size,
                              hipStream_t stream)
{
  (void)in_sizes; (void)n_in; (void)out_size; (void)ws_size;
  const int N0 = 8192, N1 = 2048, N2 = 512;
  const float R0sq = 0.05f*0.05f, R1sq = 0.10f*0.10f;

  const float* x   = (const float*)d_in[0];
  const float* pos = (const float*)d_in[1];
  // d_in[2] = batch (single graph) - unused
  int p = 3;
  const float *clsW0=(const float*)d_in[p++], *clsb0=(const float*)d_in[p++];
  const float *clsW1=(const float*)d_in[p++], *clsb1=(const float*)d_in[p++];
  const float *d00W=(const float*)d_in[p++], *d00b=(const float*)d_in[p++],
              *d00be=(const float*)d_in[p++], *d00g=(const float*)d_in[p++];
  const float *d01W=(const float*)d_in[p++], *d01b=(const float*)d_in[p++],
              *d01be=(const float*)d_in[p++], *d01g=(const float*)d_in[p++];
  const float *d10W=(const float*)d_in[p++], *d10b=(const float*)d_in[p++],
              *d10be=(const float*)d_in[p++], *d10g=(const float*)d_in[p++];
  const float *d11W=(const float*)d_in[p++], *d11b=(const float*)d_in[p++],
              *d11be=(const float*)d_in[p++], *d11g=(const float*)d_in[p++];
  const float *smW=(const float*)d_in[p++],  *smb=(const float*)d_in[p++];
  const float *u0W=(const float*)d_in[p++],  *u0b=(const float*)d_in[p++],
              *u0be=(const float*)d_in[p++], *u0g=(const float*)d_in[p++];
  const float *u1W=(const float*)d_in[p++],  *u1b=(const float*)d_in[p++],
              *u1be=(const float*)d_in[p++], *u1g=(const float*)d_in[p++];

  // -------- workspace bump allocator (d_ws is 256B aligned) --------
  char* base = (char*)d_ws; size_t off = 0;
  auto alloc = [&](size_t bytes) -> void* {
    void* r = base + off;
    off = (off + bytes + 255) & ~(size_t)255;
    return r;
  };
  int*   idx0 = (int*)  alloc((size_t)N0*KNN*4);
  int*   msk0 = (int*)  alloc((size_t)N0*KNN*4);
  int*   idx1 = (int*)  alloc((size_t)N1*KNN*4);
  int*   msk1 = (int*)  alloc((size_t)N1*KNN*4);
  int*   fpi0 = (int*)  alloc((size_t)N1*4);
  int*   fpi1 = (int*)  alloc((size_t)N2*4);
  float* stats= (float*)alloc(1024*4);
  float* x0   = (float*)alloc((size_t)N0*128*4);
  float* x0s  = (float*)alloc((size_t)N1*128*4);
  float* pos1 = (float*)alloc((size_t)N1*3*4);
  float* x1   = (float*)alloc((size_t)N1*256*4);
  float* pos2 = (float*)alloc((size_t)N2*3*4);
  float* x1s  = (float*)alloc((size_t)N2*256*4);
  // transposed weights [N,Kd]
  float* d00Wt = (float*)alloc((size_t)7*64*4);
  float* d01Wt = (float*)alloc((size_t)64*128*4);
  float* d10Wt = (float*)alloc((size_t)131*128*4);
  float* d11Wt = (float*)alloc((size_t)128*256*4);
  float* smWt  = (float*)alloc((size_t)256*256*4);
  float* u0Wt  = (float*)alloc((size_t)512*256*4);
  float* u1Wt  = (float*)alloc((size_t)384*128*4);
  float* clsW0t= (float*)alloc((size_t)128*128*4);
  float* clsW1t= (float*)alloc((size_t)128*16*4);
  float* bigA = (float*)alloc((size_t)8585216*4);   // feat0 [262144,7] / feat1 [65536,131]
  float* bigB = (float*)alloc((size_t)16777216*4);  // h0a [262144,64] / h1a [65536,128]
  float* bigC = (float*)alloc((size_t)33554432*4);  // h0b [262144,128] / h1b [65536,256]
  // up-path buffers alias into bigC (level-0/1 hiddens are dead by then)
  float* hsum    = bigC;                            // [512,256]
  float* interp1 = hsum    + (size_t)N2*256;        // [2048,256]
  float* cat0    = interp1 + (size_t)N1*256;        // [2048,512]
  float* u0h     = cat0    + (size_t)N1*512;        // [2048,256]
  float* interp0 = u0h     + (size_t)N1*256;        // [8192,256]
  float* cat1    = interp0 + (size_t)N0*256;        // [8192,384]
  float* u1h     = cat1    + (size_t)N0*384;        // [8192,128]
  float* c0h     = u1h     + (size_t)N0*128;        // [8192,128]

  auto transpose = [&](const float* W, int Kd, int Nc, float* Wt) {
    k_transpose<<<cdivl((long)Kd*Nc, 256), 256, 0, stream>>>(W, Kd, Nc, Wt);
  };
  auto gemm = [&](const float* A, const float* Wt, const float* b, float* Cm,
                  long M, int Kd, int Nc, int relu) {
    long tiles = (M >> 4) * (long)(Nc >> 4);
    k_gemm_wmma<<<(int)((tiles + 7) >> 3), 256, 0, stream>>>(A, Wt, b, Cm, (int)M, Kd, Nc, relu);
  };
  auto bn = [&](float* h, const int* mask, long Mr, int Cc,
                const float* g, const float* be) {
    k_bn_stats<<<Cc, 256, 0, stream>>>(h, mask, Mr, Cc, stats);
    k_bn_apply_relu<<<cdivl(Mr*Cc, 256), 256, 0, stream>>>(h, Mr, Cc, stats, g, be);
  };

  // ---- one-shot weight transposes (tiny) ----
  transpose(d00W,   7,  64, d00Wt);
  transpose(d01W,  64, 128, d01Wt);
  transpose(d10W, 131, 128, d10Wt);
  transpose(d11W, 128, 256, d11Wt);
  transpose(smW,  256, 256, smWt);
  transpose(u0W,  512, 256, u0Wt);
  transpose(u1W,  384, 128, u1Wt);
  transpose(clsW0,128, 128, clsW0t);
  transpose(clsW1,128,  16, clsW1t);

  // ---- SA level 0: radius graph + PointNetConv [7->64->128], max agg ----
  k_radius_topk<<<cdivl(N0,64), 64, (size_t)64*KNN*8, stream>>>(pos, N0, R0sq, idx0, msk0);
  k_build_feat<<<cdivl((long)N0*KNN,256), 256, 0, stream>>>(x, 4, pos, idx0, N0, bigA);
  gemm(bigA, d00Wt, d00b, bigB, (long)N0*KNN, 7, 64, 0);
  bn(bigB, msk0, (long)N0*KNN, 64, d00g, d00be);
  gemm(bigB, d01Wt, d01b, bigC, (long)N0*KNN, 64, 128, 0);
  bn(bigC, msk0, (long)N0*KNN, 128, d01g, d01be);
  k_max_agg<<<cdivl((long)N0*128,256), 256, 0, stream>>>(bigC, msk0, N0, 128, x0);

  // ---- FPS 8192 -> 2048, gather ----
  k_fps<<<1, 1024, (size_t)N0*4, stream>>>(pos, N0, N1, fpi0);
  k_gather_rows<<<cdivl((long)N1*3,256),   256, 0, stream>>>(pos, 3,   fpi0, N1, pos1);
  k_gather_rows<<<cdivl((long)N1*128,256), 256, 0, stream>>>(x0, 128, fpi0, N1, x0s);

  // ---- SA level 1: [131->128->256], max agg ----
  k_radius_topk<<<cdivl(N1,64), 64, (size_t)64*KNN*8, stream>>>(pos1, N1, R1sq, idx1, msk1);
  k_build_feat<<<cdivl((long)N1*KNN,256), 256, 0, stream>>>(x0s, 128, pos1, idx1, N1, bigA);
  gemm(bigA, d10Wt, d10b, bigB, (long)N1*KNN, 131, 128, 0);
  bn(bigB, msk1, (long)N1*KNN, 128, d10g, d10be);
  gemm(bigB, d11Wt, d11b, bigC, (long)N1*KNN, 128, 256, 0);
  bn(bigC, msk1, (long)N1*KNN, 256, d11g, d11be);
  k_max_agg<<<cdivl((long)N1*256,256), 256, 0, stream>>>(bigC, msk1, N1, 256, x1);

  // ---- FPS 2048 -> 512, gather ----
  k_fps<<<1, 1024, (size_t)N1*4, stream>>>(pos1, N1, N2, fpi1);
  k_gather_rows<<<cdivl((long)N2*3,256),   256, 0, stream>>>(pos1, 3,   fpi1, N2, pos2);
  k_gather_rows<<<cdivl((long)N2*256,256), 256, 0, stream>>>(x1, 256, fpi1, N2, x1s);

  // ---- summit MLP [256->256] (linear+relu, no norm) ----
  gemm(x1s, smWt, smb, hsum, N2, 256, 256, 1);

  // ---- FP up0: knn3 interp (512->2048), concat skip, [512->256]+BN+relu ----
  k_knn3_interp<<<cdivl(N1,128), 128, 0, stream>>>(pos1, N1, pos2, N2, hsum, 256, interp1);
  k_concat2<<<cdivl((long)N1*512,256), 256, 0, stream>>>(interp1, 256, x1, 256, N1, cat0);
  gemm(cat0, u0Wt, u0b, u0h, N1, 512, 256, 0);
  bn(u0h, nullptr, N1, 256, u0g, u0be);

  // ---- FP up1: knn3 interp (2048->8192), concat skip, [384->128]+BN+relu ----
  k_knn3_interp<<<cdivl(N0,128), 128, 0, stream>>>(pos, N0, pos1, N1, u0h, 256, interp0);
  k_concat2<<<cdivl((long)N0*384,256), 256, 0, stream>>>(interp0, 256, x0, 128, N0, cat1);
  gemm(cat1, u1Wt, u1b, u1h, N0, 384, 128, 0);
  bn(u1h, nullptr, N0, 128, u1g, u1be);

  // ---- classifier: [128->128]+relu, [128->16] plain ----
  gemm(u1h, clsW0t, clsb0, c0h, N0, 128, 128, 1);
  gemm(c0h, clsW1t, clsb1, (float*)d_out, N0, 128, 16, 0);
}